// GLALayer_62302795596486
// MI455X (gfx1250) — compile-verified
//
#include <hip/hip_runtime.h>
#include <hip/hip_bf16.h>

#define D_MODEL 1024
#define CHUNK   64

typedef __attribute__((ext_vector_type(2))) float v2f;
typedef __attribute__((ext_vector_type(8))) float v8f;

// ---- CDNA5 async Global->LDS copy (ASYNCcnt) with safe fallback -----------
#if defined(__has_builtin)
#if __has_builtin(__builtin_amdgcn_global_load_async_to_lds_b128) && \
    __has_builtin(__builtin_amdgcn_s_wait_asynccnt)
#define USE_ASYNC_COPY 1
#endif
#endif
#ifndef USE_ASYNC_COPY
#define USE_ASYNC_COPY 0
#endif

typedef int v4i __attribute__((vector_size(16)));
typedef __attribute__((address_space(3))) v4i lds_v4i;

// Copy 16 bytes global -> LDS. Async path bypasses VGPRs (tracked by ASYNCcnt).
static __device__ __forceinline__ void copy16(const float* __restrict__ g,
                                              float* l) {
#if USE_ASYNC_COPY
  // Generic shared pointers carry the LDS byte offset in their low 32 bits;
  // as(3) pointers are 32-bit, so the integer cast keeps exactly that offset.
  __builtin_amdgcn_global_load_async_to_lds_b128(
      (v4i*)(uintptr_t)g,
      (lds_v4i*)(unsigned)(uintptr_t)l,
      0, 0);
#else
  *(float4*)l = *(const float4*)g;
#endif
}
static __device__ __forceinline__ void copy_fence() {
#if USE_ASYNC_COPY
  __builtin_amdgcn_s_wait_asynccnt(0);   // s_wait_asynccnt 0
#endif
}

static __device__ __forceinline__ v8f wmma4(v2f a, v2f b, v8f c) {
  // D = A(16x4) * B(4x16) + C(16x16), fp32 throughout.
  return __builtin_amdgcn_wmma_f32_16x16x4_f32(false, a, false, b, (short)0, c,
                                               false, false);
}

// ---------------------------------------------------------------------------
// C[m,n] = epi( sum_k A[m,k] * B[n,k] )   ("NT" GEMM, both operands row-major)
// epi: 0 = none, 1 = sigmoid, 2 = += Res[m,n]
// Block tile 128x128, BK=16, double-buffered LDS ping-pong:
//   iter t:  fence+barrier (publish buf[cur]) -> issue async copies buf[cur^1]
//            -> WMMA on buf[cur]  (DMA overlaps compute; 1 barrier/tile)
// 8 waves 4(m) x 2(n); wave = 32x64 region = 2x4 WMMA tiles.
// LDS row stride 20 floats = 80B: 16B-aligned + 20*m mod 64 distinct.
// ---------------------------------------------------------------------------
__global__ __launch_bounds__(256)
void gemm_nt_wmma(const float* __restrict__ A, const float* __restrict__ Bm,
                  const float* __restrict__ Res, float* __restrict__ Cm,
                  int Kdim, int Ndim, int epi)
{
  __shared__ __align__(16) float As[2][128][20];
  __shared__ __align__(16) float Bs[2][128][20];

  const int tid  = threadIdx.x;
  const int lane = tid & 31;
  const int wid  = tid >> 5;
  const int lm   = lane & 15;     // M/N index within a 16-wide tile
  const int lh   = lane >> 4;     // lane half: K pair (A/B) or +8 rows (C/D)
  const int wm   = (wid & 3) * 32;
  const int wn   = (wid >> 2) * 64;
  const size_t row0 = (size_t)blockIdx.y * 128;
  const size_t col0 = (size_t)blockIdx.x * 128;

  auto stage = [&](int k0, int buf) {
#pragma unroll
    for (int i = tid; i < 512; i += 256) {        // 128x16 = 512 float4 / operand
      const int r = i >> 2, c4 = (i & 3) * 4;
      copy16(&A[(row0 + r) * (size_t)Kdim + k0 + c4], &As[buf][r][c4]);
      copy16(&Bm[(col0 + r) * (size_t)Kdim + k0 + c4], &Bs[buf][r][c4]);
    }
  };

  v8f acc[2][4] = {};

  stage(0, 0);                     // prologue: tile 0 in flight
  const int nT = Kdim / 16;
  for (int t = 0; t < nT; ++t) {
    const int cur = t & 1;
    copy_fence();                  // own async copies for buf[cur] landed
    __syncthreads();               // publish buf[cur]; orders WAR on buf[cur^1]
    if (t + 1 < nT) stage((t + 1) * 16, cur ^ 1);   // DMA next tile under compute
#pragma unroll
    for (int ks = 0; ks < 16; ks += 4) {
      v2f af[2], bf[4];
#pragma unroll
      for (int mt = 0; mt < 2; ++mt) {
        const int m = wm + mt * 16 + lm;
        af[mt].x = As[cur][m][ks + 2 * lh];
        af[mt].y = As[cur][m][ks + 2 * lh + 1];
      }
#pragma unroll
      for (int nt = 0; nt < 4; ++nt) {
        const int n = wn + nt * 16 + lm;
        bf[nt].x = Bs[cur][n][ks + 2 * lh];
        bf[nt].y = Bs[cur][n][ks + 2 * lh + 1];
      }
#pragma unroll
      for (int mt = 0; mt < 2; ++mt)
#pragma unroll
        for (int nt = 0; nt < 4; ++nt)
          acc[mt][nt] = wmma4(af[mt], bf[nt], acc[mt][nt]);
    }
  }

  // Epilogue. C/D layout: VGPR v holds row (v + 8*laneHalf), col = lane%16.
#pragma unroll
  for (int mt = 0; mt < 2; ++mt)
#pragma unroll
    for (int nt = 0; nt < 4; ++nt)
#pragma unroll
      for (int v = 0; v < 8; ++v) {
        const size_t gm = row0 + wm + mt * 16 + v + 8 * lh;
        const size_t gn = col0 + wn + nt * 16 + lm;
        float val = acc[mt][nt][v];
        if (epi == 1)      val = 1.0f / (1.0f + __expf(-val));
        else if (epi == 2) val += Res[gm * (size_t)Ndim + gn];
        Cm[gm * (size_t)Ndim + gn] = val;
      }
}

// ---------------------------------------------------------------------------
// Per-chunk masked score: S[c] = M ∘ (Qc · Kcᵀ), 64x64 per chunk, K=1024.
// One workgroup per chunk; same double-buffered ping-pong, BK=32.
// Waves 2(m,32 rows) x 4(n,16 cols). Row stride 36 floats = 144B.
// ---------------------------------------------------------------------------
__global__ __launch_bounds__(256)
void chunk_s_wmma(const float* __restrict__ Q, const float* __restrict__ Kp,
                  const float* __restrict__ log_decay, float* __restrict__ S)
{
  __shared__ __align__(16) float Qs[2][64][36];
  __shared__ __align__(16) float Ks[2][64][36];
  const int tid = threadIdx.x, lane = tid & 31, wid = tid >> 5;
  const int lm = lane & 15, lh = lane >> 4;
  const int wm = (wid & 1) * 32;
  const int wn = (wid >> 1) * 16;
  const size_t row0 = (size_t)blockIdx.x * CHUNK;

  auto stage = [&](int k0, int buf) {
#pragma unroll
    for (int i = tid; i < 512; i += 256) {        // 64x32 = 512 float4 / operand
      const int r = i >> 3, c4 = (i & 7) * 4;
      copy16(&Q[(row0 + r) * D_MODEL + k0 + c4], &Qs[buf][r][c4]);
      copy16(&Kp[(row0 + r) * D_MODEL + k0 + c4], &Ks[buf][r][c4]);
    }
  };

  v8f acc[2] = {};

  stage(0, 0);
  const int nT = D_MODEL / 32;
  for (int t = 0; t < nT; ++t) {
    const int cur = t & 1;
    copy_fence();
    __syncthreads();
    if (t + 1 < nT) stage((t + 1) * 32, cur ^ 1);
#pragma unroll
    for (int ks = 0; ks < 32; ks += 4) {
      v2f bf;
      const int n = wn + lm;
      bf.x = Ks[cur][n][ks + 2 * lh];
      bf.y = Ks[cur][n][ks + 2 * lh + 1];
#pragma unroll
      for (int mt = 0; mt < 2; ++mt) {
        v2f af;
        const int m = wm + mt * 16 + lm;
        af.x = Qs[cur][m][ks + 2 * lh];
        af.y = Qs[cur][m][ks + 2 * lh + 1];
        acc[mt] = wmma4(af, bf, acc[mt]);
      }
    }
  }

  const float dec = 1.0f / (1.0f + __expf(-log_decay[0]));
  const float l2d = __log2f(dec);
  float* Sc = S + (size_t)blockIdx.x * (CHUNK * CHUNK);
#pragma unroll
  for (int mt = 0; mt < 2; ++mt)
#pragma unroll
    for (int v = 0; v < 8; ++v) {
      const int i = wm + mt * 16 + v + 8 * lh;
      const int j = wn + lm;
      const int diff = i - j;
      const float msk = (diff >= 0) ? exp2f(l2d * (float)diff) : 0.0f;
      Sc[i * CHUNK + j] = acc[mt][v] * msk;
    }
}

// ---------------------------------------------------------------------------
// GO = gate ∘ (S · Vc).  Grid: (n-blocks of 256, chunk).  S cached in LDS,
// V streamed in 16x256 slices. Waves 2(m,32 rows) x 4(n,64 cols).
// Ss row stride 68 floats = 272B: 16B-aligned, 68*m mod 64 = 4m distinct.
// ---------------------------------------------------------------------------
__global__ __launch_bounds__(256)
void chunk_o_wmma(const float* __restrict__ S, const float* __restrict__ V,
                  const float* __restrict__ G, float* __restrict__ GO)
{
  __shared__ __align__(16) float Ss[64][68];
  __shared__ __align__(16) float Vs[16][256];
  const int tid = threadIdx.x, lane = tid & 31, wid = tid >> 5;
  const int lm = lane & 15, lh = lane >> 4;
  const int wm = (wid & 1) * 32;
  const int wn = (wid >> 1) * 64;
  const size_t row0 = (size_t)blockIdx.y * CHUNK;
  const int n0 = blockIdx.x * 256;

  const float* Sc = S + (size_t)blockIdx.y * (CHUNK * CHUNK);
#pragma unroll
  for (int i = tid; i < 1024; i += 256) {         // 64x64 S tile = 1024 float4
    const int r = i >> 4, c4 = (i & 15) * 4;
    copy16(&Sc[r * CHUNK + c4], &Ss[r][c4]);
  }

  v8f acc[2][4] = {};

  for (int kk = 0; kk < CHUNK; kk += 16) {
    __syncthreads();   // previous Vs fully consumed
#pragma unroll
    for (int i = tid; i < 1024; i += 256) {       // 16x256 V slice = 1024 float4
      const int r = i >> 6, c4 = (i & 63) * 4;
      copy16(&V[(row0 + kk + r) * D_MODEL + n0 + c4], &Vs[r][c4]);
    }
    copy_fence();      // Ss (first iter) + Vs async copies done
    __syncthreads();   // publish
#pragma unroll
    for (int ks = 0; ks < 16; ks += 4) {
      v2f af[2], bf[4];
#pragma unroll
      for (int mt = 0; mt < 2; ++mt) {
        const int m = wm + mt * 16 + lm;
        af[mt].x = Ss[m][kk + ks + 2 * lh];
        af[mt].y = Ss[m][kk + ks + 2 * lh + 1];
      }
#pragma unroll
      for (int nt = 0; nt < 4; ++nt) {
        const int n = wn + nt * 16 + lm;
        bf[nt].x = Vs[ks + 2 * lh][n];
        bf[nt].y = Vs[ks + 2 * lh + 1][n];
      }
#pragma unroll
      for (int mt = 0; mt < 2; ++mt)
#pragma unroll
        for (int nt = 0; nt < 4; ++nt)
          acc[mt][nt] = wmma4(af[mt], bf[nt], acc[mt][nt]);
    }
  }

#pragma unroll
  for (int mt = 0; mt < 2; ++mt)
#pragma unroll
    for (int nt = 0; nt < 4; ++nt)
#pragma unroll
      for (int v = 0; v < 8; ++v) {
        const size_t gm = row0 + wm + mt * 16 + v + 8 * lh;
        const size_t gn = (size_t)n0 + wn + nt * 16 + lm;
        GO[gm * D_MODEL + gn] = G[gm * D_MODEL + gn] * acc[mt][nt][v];
      }
}

// ---------------------------------------------------------------------------
extern "C" void kernel_launch(void* const* d_in, const int* in_sizes, int n_in,
                              void* d_out, int out_size, void* d_ws, size_t ws_size,
                              hipStream_t stream)
{
  (void)in_sizes; (void)n_in; (void)out_size; (void)ws_size;

  const float* x  = (const float*)d_in[0];
  const float* Wq = (const float*)d_in[1];
  const float* Wk = (const float*)d_in[2];
  const float* Wv = (const float*)d_in[3];
  const float* Wg = (const float*)d_in[4];
  const float* Wo = (const float*)d_in[5];
  const float* ld = (const float*)d_in[6];
  float* out = (float*)d_out;

  const int M = 4 * 8192;                       // B*T rows
  const size_t MD = (size_t)M * D_MODEL;
  const int nchunks = M / CHUNK;                // 512

  float* Q  = (float*)d_ws;
  float* Kk = Q  + MD;
  float* Vv = Kk + MD;
  float* G  = Vv + MD;
  float* S  = G  + MD;
  float* GO = S  + (size_t)nchunks * CHUNK * CHUNK;

  dim3 gg(D_MODEL / 128, M / 128);
  gemm_nt_wmma<<<gg, 256, 0, stream>>>(x, Wq, nullptr, Q,  D_MODEL, D_MODEL, 0);
  gemm_nt_wmma<<<gg, 256, 0, stream>>>(x, Wk, nullptr, Kk, D_MODEL, D_MODEL, 0);
  gemm_nt_wmma<<<gg, 256, 0, stream>>>(x, Wv, nullptr, Vv, D_MODEL, D_MODEL, 0);
  gemm_nt_wmma<<<gg, 256, 0, stream>>>(x, Wg, nullptr, G,  D_MODEL, D_MODEL, 1);

  chunk_s_wmma<<<nchunks, 256, 0, stream>>>(Q, Kk, ld, S);
  chunk_o_wmma<<<dim3(D_MODEL / 256, nchunks), 256, 0, stream>>>(S, Vv, G, GO);

  gemm_nt_wmma<<<gg, 256, 0, stream>>>(GO, Wo, x, out, D_MODEL, D_MODEL, 2);
}